// BiLSTM_CRF_47184510713942
// MI455X (gfx1250) — compile-verified
//
#include <hip/hip_runtime.h>
#include <cstdint>
#include <cstddef>

// ---- dimensions (fixed by the reference) ----
#define V_   32000
#define E_   200
#define HD_  256
#define T_   9
#define B_   128
#define S_   512
#define H_   128      // HD/2
#define G_   512      // 4*H
#define XPAD 224      // padded K for layer-0 (7 chunks of 32, zero-filled tail)

typedef __attribute__((ext_vector_type(2))) float v2f;
typedef __attribute__((ext_vector_type(8))) float v8f;

// async-to-LDS builtin wants int4* in AS(1)/AS(3)
typedef int v4i __attribute__((vector_size(16)));
typedef __attribute__((address_space(1))) v4i as1_v4i;
typedef __attribute__((address_space(3))) v4i as3_v4i;

__device__ __forceinline__ v8f wmma_f32_k4(v2f a, v2f b, v8f c) {
    // D = A(16x4 f32) * B(4x16 f32) + C(16x16 f32)
    return __builtin_amdgcn_wmma_f32_16x16x4_f32(false, a, false, b, (short)0, c, false, false);
}

__device__ __forceinline__ float sigm(float x) { return 1.0f / (1.0f + expf(-x)); }

// ---------------------------------------------------------------------------
// 1) Embedding gather with padded row stride:
//    x[s][b][e] = emb[sentence[b][s]][e]  for e<200, 0 for 200<=e<224
// ---------------------------------------------------------------------------
__global__ __launch_bounds__(256) void embed_kernel(const int* __restrict__ sent,
                                                    const float* __restrict__ emb,
                                                    float* __restrict__ x) {
    size_t i = (size_t)blockIdx.x * 256 + threadIdx.x;
    const size_t total = (size_t)S_ * B_ * XPAD;
    if (i >= total) return;
    int e = (int)(i % XPAD);
    size_t r = i / XPAD;
    int b = (int)(r % B_);
    int s = (int)(r / B_);
    float v = 0.0f;
    if (e < E_) {
        int tok = sent[(size_t)b * S_ + s];
        v = emb[(size_t)tok * E_ + e];
    }
    x[i] = v;
}

// ---------------------------------------------------------------------------
// 1b) Zero-pad layer-0 input weights: (2,512,200) -> (2,512,224)
// ---------------------------------------------------------------------------
__global__ __launch_bounds__(256) void pack_w_kernel(const float* __restrict__ w,
                                                     float* __restrict__ wp) {
    size_t i = (size_t)blockIdx.x * 256 + threadIdx.x;
    const size_t total = (size_t)2 * G_ * XPAD;
    if (i >= total) return;
    int k = (int)(i % XPAD);
    size_t n = i / XPAD;
    wp[i] = (k < E_) ? w[n * E_ + k] : 0.0f;
}

// ---------------------------------------------------------------------------
// 2) GEMM: Y(Mx512) = X(MxK)*W(512xK)^T + (b1[n]+b2[n])  via f32 WMMA 16x16x4
//    Tiles of X and W are staged into LDS with async-to-LDS bulk copies,
//    double-buffered on ASYNCcnt; WMMA fragments come from conflict-free LDS.
//    grid = (M/128, 512/128), block = 256 (8 waves). K consumed in 32-chunks.
// ---------------------------------------------------------------------------
#define KC       32
#define LSTRIDE  36                       // floats per LDS tile row (16B aligned, bank-clean)
#define TILE_FL  (128 * LSTRIDE)          // floats per tile buffer
#define TILE_BY  (TILE_FL * 4)            // 18432 bytes

__device__ __forceinline__ void stage_tile_async(const float* src, int strideF,
                                                 uint32_t ldsByteBase, int tid) {
    // copy a 128-row x 32-float tile; 1024 16B chunks over 256 threads
#pragma unroll
    for (int i = 0; i < 4; ++i) {
        int idx = tid + i * 256;
        int row = idx >> 3;
        int c16 = idx & 7;
        const float* g = src + (size_t)row * strideF + c16 * 4;
        uint32_t loff = ldsByteBase + (uint32_t)row * (LSTRIDE * 4) + (uint32_t)c16 * 16;
        __builtin_amdgcn_global_load_async_to_lds_b128(
            (as1_v4i*)(uintptr_t)g,
            (as3_v4i*)(uintptr_t)loff,
            0, 0);
    }
}

__global__ __launch_bounds__(256) void gemm_xp_kernel(const float* __restrict__ X, int xsF,
                                                      const float* __restrict__ W, int wsF,
                                                      int nchunks,
                                                      const float* __restrict__ b1,
                                                      const float* __restrict__ b2,
                                                      float* __restrict__ Y) {
    extern __shared__ float smem[];                   // [4 * TILE_FL] : X0 X1 W0 W1
    const uint32_t smemBase = (uint32_t)(uintptr_t)smem;  // LDS byte offset (aperture low bits)

    const int tid  = threadIdx.x;
    const int lane = tid & 31;
    const int wv   = tid >> 5;
    const int l16  = lane & 15;
    const int hi   = lane >> 4;

    const int mblk = blockIdx.x * 128;
    const int n0   = blockIdx.y * 128;

    const float* Xt = X + (size_t)mblk * xsF;
    const float* Wt = W + (size_t)n0 * wsF;

    v8f acc[8] = {};

    // prime chunk 0 into buffers {X0, W0}
    stage_tile_async(Xt, xsF, smemBase + 0 * TILE_BY, tid);
    stage_tile_async(Wt, wsF, smemBase + 2 * TILE_BY, tid);

    for (int c = 0; c < nchunks; ++c) {
        const int cur = c & 1;
        if (c + 1 < nchunks) {
            const int nxt = 1 - cur;
            stage_tile_async(Xt + (size_t)(c + 1) * KC, xsF, smemBase + (uint32_t)nxt * TILE_BY, tid);
            stage_tile_async(Wt + (size_t)(c + 1) * KC, wsF, smemBase + (uint32_t)(2 + nxt) * TILE_BY, tid);
            __builtin_amdgcn_s_wait_asynccnt(8);      // chunk c fully landed, c+1 in flight
        } else {
            __builtin_amdgcn_s_wait_asynccnt(0);
        }
        __syncthreads();

        const float* xb = smem + (size_t)cur * TILE_FL;
        const float* wb = smem + (size_t)(2 + cur) * TILE_FL;
        const int arow = (wv * 16 + l16) * LSTRIDE + 2 * hi;

#pragma unroll
        for (int k0 = 0; k0 < KC; k0 += 4) {
            v2f a = *(const v2f*)(xb + arow + k0);
#pragma unroll
            for (int j = 0; j < 8; ++j) {
                v2f bf = *(const v2f*)(wb + (j * 16 + l16) * LSTRIDE + 2 * hi + k0);
                acc[j] = wmma_f32_k4(a, bf, acc[j]);
            }
        }
        __syncthreads();   // all waves done reading buf[cur] before it is restaged
    }

#pragma unroll
    for (int j = 0; j < 8; ++j) {
        int n = n0 + j * 16 + l16;
        float bias = b1[n] + b2[n];
#pragma unroll
        for (int r = 0; r < 8; ++r) {
            int m = mblk + wv * 16 + r + 8 * hi;
            Y[(size_t)m * G_ + n] = acc[j][r] + bias;
        }
    }
}

// ---------------------------------------------------------------------------
// 3) LSTM recurrence (one direction per block; block = 1024 thr = 32 waves).
//    g = xp[t] + h @ whh^T ; gates -> c,h. h kept transposed in LDS (padded
//    stride). Fragment loads are software-pipelined one k-step ahead so WMMA
//    doesn't sit on zero-distance waitcnts.
// ---------------------------------------------------------------------------
#define BSTRIDE 129   // padded row stride of hT (floats)

__global__ __launch_bounds__(1024) void lstm_recur_kernel(const float* __restrict__ xp0,
                                                          const float* __restrict__ xp1,
                                                          const float* __restrict__ whh_all, // (2,4H,H)
                                                          float* __restrict__ hout) {       // (S,B,2H)
    extern __shared__ float hT[];      // H_ * BSTRIDE floats

    const int d      = blockIdx.x;     // direction
    const float* xp  = d ? xp1 : xp0;
    const float* whh = whh_all + (size_t)d * G_ * H_;

    const int tid  = threadIdx.x;
    const int lane = tid & 31;
    const int wv   = tid >> 5;         // 0..31
    const int l16  = lane & 15;
    const int hi   = lane >> 4;

    for (int i = tid; i < H_ * BSTRIDE; i += 1024) hT[i] = 0.0f;

    v8f cstate[2] = {};
    v8f hsave[2];

    __syncthreads();

    for (int s = 0; s < S_; ++s) {
        const int t = d ? (S_ - 1 - s) : s;
        const float* xpt = xp + (size_t)t * B_ * G_;

#pragma unroll
        for (int ti = 0; ti < 2; ++ti) {
            const int task = wv + ti * 32;      // 0..63
            const int m0 = (task >> 3) * 16;    // batch-tile origin
            const int j0 = (task & 7) * 16;     // hidden-tile origin

            v8f g0, g1, g2, g3;
#pragma unroll
            for (int r = 0; r < 8; ++r) {
                const float* pr = xpt + (size_t)(m0 + r + 8 * hi) * G_ + l16;
                g0[r] = pr[j0];
                g1[r] = pr[128 + j0];
                g2[r] = pr[256 + j0];
                g3[r] = pr[384 + j0];
            }

            const float* wr0 = whh + 2 * hi;
            const int aoff = m0 + l16;

            // prologue: fragments for k0 = 0
            v2f a, b0, b1, b2, b3;
            a.x = hT[(2 * hi) * BSTRIDE + aoff];
            a.y = hT[(1 + 2 * hi) * BSTRIDE + aoff];
            b0 = *(const v2f*)(wr0 + (size_t)(j0 + l16) * H_);
            b1 = *(const v2f*)(wr0 + (size_t)(128 + j0 + l16) * H_);
            b2 = *(const v2f*)(wr0 + (size_t)(256 + j0 + l16) * H_);
            b3 = *(const v2f*)(wr0 + (size_t)(384 + j0 + l16) * H_);

            for (int k0 = 0; k0 < H_ - 4; k0 += 4) {
                const int kn = k0 + 4;
                v2f an, bn0, bn1, bn2, bn3;
                an.x = hT[(kn + 2 * hi) * BSTRIDE + aoff];
                an.y = hT[(kn + 1 + 2 * hi) * BSTRIDE + aoff];
                const float* wr = wr0 + kn;
                bn0 = *(const v2f*)(wr + (size_t)(j0 + l16) * H_);
                bn1 = *(const v2f*)(wr + (size_t)(128 + j0 + l16) * H_);
                bn2 = *(const v2f*)(wr + (size_t)(256 + j0 + l16) * H_);
                bn3 = *(const v2f*)(wr + (size_t)(384 + j0 + l16) * H_);

                g0 = wmma_f32_k4(a, b0, g0);
                g1 = wmma_f32_k4(a, b1, g1);
                g2 = wmma_f32_k4(a, b2, g2);
                g3 = wmma_f32_k4(a, b3, g3);

                a = an; b0 = bn0; b1 = bn1; b2 = bn2; b3 = bn3;
            }
            // epilogue: last k-step
            g0 = wmma_f32_k4(a, b0, g0);
            g1 = wmma_f32_k4(a, b1, g1);
            g2 = wmma_f32_k4(a, b2, g2);
            g3 = wmma_f32_k4(a, b3, g3);

#pragma unroll
            for (int r = 0; r < 8; ++r) {
                float iv = sigm(g0[r]);
                float fv = sigm(g1[r]);
                float gv = tanhf(g2[r]);
                float ov = sigm(g3[r]);
                float cv = fv * cstate[ti][r] + iv * gv;
                cstate[ti][r] = cv;
                hsave[ti][r]  = ov * tanhf(cv);
            }
        }

        __syncthreads();   // all reads of hT done before overwrite

#pragma unroll
        for (int ti = 0; ti < 2; ++ti) {
            const int task = wv + ti * 32;
            const int m0 = (task >> 3) * 16;
            const int j0 = (task & 7) * 16;
            const int j  = j0 + l16;
#pragma unroll
            for (int r = 0; r < 8; ++r) {
                const int b = m0 + r + 8 * hi;
                hT[j * BSTRIDE + b] = hsave[ti][r];
                hout[((size_t)t * B_ + b) * HD_ + d * H_ + j] = hsave[ti][r];
            }
        }

        __syncthreads();   // hT fully updated before next step's reads
    }
}

// ---------------------------------------------------------------------------
// 4) Emissions: em[b][s][t] = dot(h2[s][b][:], proj_w[t][:]) + proj_b[t]
// ---------------------------------------------------------------------------
__global__ __launch_bounds__(256) void emis_kernel(const float* __restrict__ h2,
                                                   const float* __restrict__ pw,
                                                   const float* __restrict__ pb,
                                                   float* __restrict__ em) {
    int idx = blockIdx.x * 256 + threadIdx.x;     // idx = s*B + b,  0..65535
    int s = idx >> 7;
    int b = idx & 127;
    const float* hp = h2 + (size_t)idx * HD_;
    float acc[T_] = {};
    for (int k = 0; k < HD_; ++k) {
        float hv = hp[k];
#pragma unroll
        for (int t = 0; t < T_; ++t) acc[t] += hv * pw[t * HD_ + k];
    }
    float* o = em + ((size_t)b * S_ + s) * T_;
#pragma unroll
    for (int t = 0; t < T_; ++t) o[t] = acc[t] + pb[t];
}

// ---------------------------------------------------------------------------
// 5) CRF NLL: one thread per sequence, exact reference semantics.
// ---------------------------------------------------------------------------
__global__ __launch_bounds__(128) void crf_kernel(const float* __restrict__ em,
                                                  const int* __restrict__ sentence,
                                                  const int* __restrict__ tags,
                                                  const float* __restrict__ st,
                                                  const float* __restrict__ et,
                                                  const float* __restrict__ tr,
                                                  float* __restrict__ out) {
    __shared__ float strans[T_], etrans[T_], T2[T_ * T_];
    __shared__ float red[B_];
    const int tid = threadIdx.x;
    if (tid < T_ * T_) T2[tid] = tr[tid];
    if (tid < T_) { strans[tid] = st[tid]; etrans[tid] = et[tid]; }
    __syncthreads();

    const int b = tid;
    const float* eb = em + (size_t)b * S_ * T_;
    const int* sb = sentence + (size_t)b * S_;
    const int* tb = tags + (size_t)b * S_;

    float alpha[T_];
    const int tag0 = tb[0];
    float num = strans[tag0] + eb[tag0];
#pragma unroll
    for (int t = 0; t < T_; ++t) alpha[t] = strans[t] + eb[t];
    int cnt = 1;   // sentence[:,0] is forced nonzero

    for (int s = 1; s < S_; ++s) {
        const bool m = (sb[s] != 0);
        if (m) {
            const int tg = tb[s];
            const float* e = eb + s * T_;
            num += T2[tb[s - 1] * T_ + tg] + e[tg];
            cnt++;
            float na[T_];
#pragma unroll
            for (int t2 = 0; t2 < T_; ++t2) {
                float mx = -3.0e38f;
#pragma unroll
                for (int t1 = 0; t1 < T_; ++t1)
                    mx = fmaxf(mx, alpha[t1] + T2[t1 * T_ + t2]);
                float ssum = 0.0f;
#pragma unroll
                for (int t1 = 0; t1 < T_; ++t1)
                    ssum += expf(alpha[t1] + T2[t1 * T_ + t2] - mx);
                na[t2] = mx + logf(ssum) + e[t2];
            }
#pragma unroll
            for (int t2 = 0; t2 < T_; ++t2) alpha[t2] = na[t2];
        }
    }

    num += etrans[tb[cnt - 1]];

    float mx = -3.0e38f;
#pragma unroll
    for (int t = 0; t < T_; ++t) mx = fmaxf(mx, alpha[t] + etrans[t]);
    float ssum = 0.0f;
#pragma unroll
    for (int t = 0; t < T_; ++t) ssum += expf(alpha[t] + etrans[t] - mx);
    const float den = mx + logf(ssum);

    red[tid] = den - num;          // loss contribution = -(num - den)
    __syncthreads();
    for (int off = 64; off > 0; off >>= 1) {
        if (tid < off) red[tid] += red[tid + off];
        __syncthreads();
    }
    if (tid == 0) out[0] = red[0];
}

// ---------------------------------------------------------------------------
// Host-side orchestration
// ---------------------------------------------------------------------------
extern "C" void kernel_launch(void* const* d_in, const int* in_sizes, int n_in,
                              void* d_out, int out_size, void* d_ws, size_t ws_size,
                              hipStream_t stream) {
    (void)in_sizes; (void)n_in; (void)out_size; (void)ws_size;

    const int*   sentence = (const int*)d_in[0];
    const int*   tags     = (const int*)d_in[1];
    const float* emb      = (const float*)d_in[2];
    const float* w_ih_l0  = (const float*)d_in[3];   // (2,512,200)
    const float* w_hh_l0  = (const float*)d_in[4];   // (2,512,128)
    const float* b_ih_l0  = (const float*)d_in[5];   // (2,512)
    const float* b_hh_l0  = (const float*)d_in[6];
    const float* w_ih_l1  = (const float*)d_in[7];   // (2,512,256)
    const float* w_hh_l1  = (const float*)d_in[8];
    const float* b_ih_l1  = (const float*)d_in[9];
    const float* b_hh_l1  = (const float*)d_in[10];
    const float* proj_w   = (const float*)d_in[11];  // (9,256)
    const float* proj_b   = (const float*)d_in[12];
    const float* st       = (const float*)d_in[13];
    const float* et       = (const float*)d_in[14];
    const float* tr       = (const float*)d_in[15];

    // workspace carve-up
    char* ws = (char*)d_ws;
    float* x    = (float*)ws;  ws += (size_t)S_ * B_ * XPAD * 4;  // 58.7 MB (padded x)
    float* wpad = (float*)ws;  ws += (size_t)2 * G_ * XPAD * 4;   // 0.9 MB  (padded w_ih_l0)
    float* xp0  = (float*)ws;  ws += (size_t)S_ * B_ * G_  * 4;   // 134 MB
    float* xp1  = (float*)ws;  ws += (size_t)S_ * B_ * G_  * 4;   // 134 MB
    float* h1   = (float*)ws;  ws += (size_t)S_ * B_ * HD_ * 4;   // 67 MB
    float* h2   = (float*)ws;  ws += (size_t)S_ * B_ * HD_ * 4;   // 67 MB
    float* emis = (float*)ws;  ws += (size_t)B_ * S_ * T_  * 4;   // 2.4 MB

    const int M = S_ * B_;   // 65536
    const size_t gemmSh = 4 * TILE_BY;               // 73728 B of LDS
    const size_t recSh  = (size_t)H_ * BSTRIDE * 4;  // ~66 KB of LDS

    // 1) embedding (padded) + weight pack (padded)
    {
        size_t total = (size_t)S_ * B_ * XPAD;
        embed_kernel<<<(int)((total + 255) / 256), 256, 0, stream>>>(sentence, emb, x);
        size_t total2 = (size_t)2 * G_ * XPAD;
        pack_w_kernel<<<(int)((total2 + 255) / 256), 256, 0, stream>>>(w_ih_l0, wpad);
    }

    // 2) layer-0 input projections (K padded to 224 = 7 chunks)
    {
        dim3 grid(M / 128, G_ / 128);
        gemm_xp_kernel<<<grid, 256, gemmSh, stream>>>(x, XPAD, wpad,                 XPAD, 7,
                                                      b_ih_l0,      b_hh_l0,      xp0);
        gemm_xp_kernel<<<grid, 256, gemmSh, stream>>>(x, XPAD, wpad + (size_t)G_ * XPAD, XPAD, 7,
                                                      b_ih_l0 + G_, b_hh_l0 + G_, xp1);
    }

    // 3) layer-0 recurrence
    lstm_recur_kernel<<<2, 1024, recSh, stream>>>(xp0, xp1, w_hh_l0, h1);

    // 4) layer-1 input projections (K = 256 = 8 chunks), reuse xp buffers
    {
        dim3 grid(M / 128, G_ / 128);
        gemm_xp_kernel<<<grid, 256, gemmSh, stream>>>(h1, HD_, w_ih_l1,                  HD_, 8,
                                                      b_ih_l1,      b_hh_l1,      xp0);
        gemm_xp_kernel<<<grid, 256, gemmSh, stream>>>(h1, HD_, w_ih_l1 + (size_t)G_ * HD_, HD_, 8,
                                                      b_ih_l1 + G_, b_hh_l1 + G_, xp1);
    }

    // 5) layer-1 recurrence
    lstm_recur_kernel<<<2, 1024, recSh, stream>>>(xp0, xp1, w_hh_l1, h2);

    // 6) emissions
    emis_kernel<<<M / 256, 256, 0, stream>>>(h2, proj_w, proj_b, emis);

    // 7) CRF NLL -> scalar
    crf_kernel<<<1, B_, 0, stream>>>(emis, sentence, tags, st, et, tr, (float*)d_out);
}